// MixtureOfExperts_90091234001165
// MI455X (gfx1250) — compile-verified
//
#include <hip/hip_runtime.h>
#include <hip/hip_bf16.h>

typedef __attribute__((ext_vector_type(2))) float v2f;
typedef __attribute__((ext_vector_type(8))) float v8f;

#define NTOK   8192
#define DIM    512
#define DFF    2048
#define NEXP   8
#define NHBLK  8        // histogram blocks (1024 tokens each)
#define MAXPAD (NTOK + NEXP * 16)   // 8320
#define MAXTIL (MAXPAD / 16)        // 520
#define SAP    516      // A-tile LDS row stride (pad to kill bank conflicts)
#define HSP    1028     // H-tile LDS row stride
#define DFC    1024     // DFF chunk held in LDS

// ws int layout: [0..7] counts, [8..15] start, [16..23] pstart,
// [24..87] bhist[8][8], [88..151] pbh[8][8], [152..152+8320) gather,
// then float outscale[8192]
#define WS_COUNTS 0
#define WS_START  8
#define WS_PSTART 16
#define WS_BHIST  24
#define WS_PBH    88
#define WS_GATHER 152
#define WS_OSCALE (WS_GATHER + MAXPAD)

__global__ void moe_hist_kernel(const int* __restrict__ idx, int* __restrict__ wsI) {
    __shared__ int h[NEXP];
    int tid = threadIdx.x;
    if (tid < NEXP) h[tid] = 0;
    __syncthreads();
    int base = blockIdx.x * 1024;
    for (int i = tid; i < 1024; i += blockDim.x)
        atomicAdd(&h[idx[base + i]], 1);
    __syncthreads();
    if (tid < NEXP) wsI[WS_BHIST + blockIdx.x * NEXP + tid] = h[tid];
}

__global__ void moe_prefix_kernel(int* __restrict__ wsI, float* __restrict__ out) {
    if (threadIdx.x == 0 && blockIdx.x == 0) {
        int run[NEXP];
        for (int e = 0; e < NEXP; ++e) run[e] = 0;
        for (int b = 0; b < NHBLK; ++b)
            for (int e = 0; e < NEXP; ++e) {
                wsI[WS_PBH + b * NEXP + e] = run[e];
                run[e] += wsI[WS_BHIST + b * NEXP + e];
            }
        int s = 0, p = 0;
        for (int e = 0; e < NEXP; ++e) {
            wsI[WS_COUNTS + e] = run[e];
            wsI[WS_START + e] = s;
            wsI[WS_PSTART + e] = p;
            s += run[e];
            p += (run[e] + 15) & ~15;
        }
        out[NTOK * DIM] = 0.0f;   // total_loss
    }
}

__global__ void moe_ginit_kernel(int* __restrict__ wsI) {
    int i = blockIdx.x * blockDim.x + threadIdx.x;
    if (i < MAXPAD) wsI[WS_GATHER + i] = -1;
}

__global__ void moe_rank_kernel(const int* __restrict__ idx,
                                const float* __restrict__ prob,
                                int* __restrict__ wsI,
                                float* __restrict__ oscale) {
    int i = blockIdx.x * blockDim.x + threadIdx.x;
    if (i >= NTOK) return;
    int a = idx[i];
    int blk = i >> 10;
    int eq = wsI[WS_PBH + blk * NEXP + a];
    for (int j = blk << 10; j < i; ++j)
        eq += (idx[j] == a) ? 1 : 0;
    int rank = wsI[WS_START + a] + eq;       // stable-sort position of token i
    oscale[i] = prob[rank];                  // out scale = prob[inv[i]]
    int ppos = wsI[WS_PSTART + a] + eq;      // position in padded gather array
    wsI[WS_GATHER + ppos] = i;
}

__device__ __forceinline__ v8f splat8(float v) {
    v8f r = {v, v, v, v, v, v, v, v};
    return r;
}

__global__ __launch_bounds__(256) void moe_gemm_kernel(
    const float* __restrict__ X, const float* __restrict__ prob,
    const float* __restrict__ W1, const float* __restrict__ b1,
    const float* __restrict__ W2, const float* __restrict__ b2,
    float* __restrict__ out, const int* __restrict__ wsI,
    const float* __restrict__ oscale) {

    __shared__ float As[16 * SAP];   // gathered x_w tile, 16 x 512
    __shared__ float Hs[16 * HSP];   // H chunk, 16 x 1024
    __shared__ int   gtile[16];
    __shared__ float sprob[16];
    __shared__ float sosc[16];
    __shared__ int   sexp;

    const int tid = threadIdx.x;
    const int tile = blockIdx.x;

    if (tid == 0) {
        int row0 = tile * 16;
        int esel = 0;
        for (int e = 0; e < NEXP; ++e) {
            int ps = wsI[WS_PSTART + e];
            int pe = ps + ((wsI[WS_COUNTS + e] + 15) & ~15);
            if (row0 >= ps && row0 < pe) esel = e;
        }
        sexp = esel;
    }
    if (tid < 16) {
        int tok = wsI[WS_GATHER + tile * 16 + tid];
        gtile[tid] = tok;
        sprob[tid] = (tok >= 0) ? prob[tok] : 0.0f;
        sosc[tid]  = (tok >= 0) ? oscale[tok] : 0.0f;
    }
    __syncthreads();
    const int e = sexp;

    // Load + gather + prescale A tile (x_w = input * prob[token])
    for (int i = tid; i < 16 * DIM; i += 256) {
        int r = i >> 9;
        int k = i & (DIM - 1);
        int tok = gtile[r];
        float v = 0.0f;
        if (tok >= 0) v = X[tok * DIM + k] * sprob[r];
        As[r * SAP + k] = v;
    }
    __syncthreads();

    const int lane = tid & 31;
    const int wave = tid >> 5;
    const int ln = lane & 15;
    const int hi = lane >> 4;

    const float* W1e = W1 + (size_t)e * DIM * DFF;
    const float* W2e = W2 + (size_t)e * DFF * DIM;
    const float* b1e = b1 + e * DFF;
    const float* b2e = b2 + e * DIM;

    // Persistent Y accumulators: each wave owns 64 output cols (4 tiles)
    v8f yacc[4];
#pragma unroll
    for (int t = 0; t < 4; ++t)
        yacc[t] = splat8(b2e[wave * 64 + t * 16 + ln]);

    for (int ch = 0; ch < DFF / DFC; ++ch) {
        const int cb = ch * DFC;

        // ---- GEMM1: H[:, cb:cb+DFC] = relu(x_w @ W1 + b1) ----
        for (int g = 0; g < 2; ++g) {
            const int wcb = cb + wave * 128 + g * 64;   // global DFF col base
            v8f hacc[4];
#pragma unroll
            for (int t = 0; t < 4; ++t)
                hacc[t] = splat8(b1e[wcb + t * 16 + ln]);

            for (int k = 0; k < DIM; k += 4) {
                const int ka = k + 2 * hi;
                v2f a = *(const v2f*)&As[ln * SAP + ka];
#pragma unroll
                for (int t = 0; t < 4; ++t) {
                    const int n = wcb + t * 16 + ln;
                    v2f bb;
                    bb.x = W1e[(size_t)ka * DFF + n];
                    bb.y = W1e[(size_t)(ka + 1) * DFF + n];
                    hacc[t] = __builtin_amdgcn_wmma_f32_16x16x4_f32(
                        false, a, false, bb, (short)0, hacc[t], false, false);
                }
            }
            const int lcb = wave * 128 + g * 64;        // local col in Hs
#pragma unroll
            for (int t = 0; t < 4; ++t)
#pragma unroll
                for (int r = 0; r < 8; ++r) {
                    float v = hacc[t][r];
                    Hs[(r + 8 * hi) * HSP + lcb + t * 16 + ln] = v > 0.0f ? v : 0.0f;
                }
        }
        __syncthreads();

        // ---- GEMM2: Y += H_chunk @ W2[cb:cb+DFC, :] ----
        for (int kk = 0; kk < DFC; kk += 4) {
            const int ka = kk + 2 * hi;
            v2f a = *(const v2f*)&Hs[ln * HSP + ka];
#pragma unroll
            for (int t = 0; t < 4; ++t) {
                const int n = wave * 64 + t * 16 + ln;
                v2f bb;
                bb.x = W2e[(size_t)(cb + ka) * DIM + n];
                bb.y = W2e[(size_t)(cb + ka + 1) * DIM + n];
                yacc[t] = __builtin_amdgcn_wmma_f32_16x16x4_f32(
                    false, a, false, bb, (short)0, yacc[t], false, false);
            }
        }
        __syncthreads();   // Hs reused next chunk
    }

    // ---- Scatter: out[token] = y * prob[rank(token)] ----
#pragma unroll
    for (int t = 0; t < 4; ++t)
#pragma unroll
        for (int r = 0; r < 8; ++r) {
            const int row = r + 8 * hi;
            const int tok = gtile[row];
            if (tok >= 0)
                out[(size_t)tok * DIM + wave * 64 + t * 16 + ln] = yacc[t][r] * sosc[row];
        }
}

extern "C" void kernel_launch(void* const* d_in, const int* in_sizes, int n_in,
                              void* d_out, int out_size, void* d_ws, size_t ws_size,
                              hipStream_t stream) {
    (void)in_sizes; (void)n_in; (void)out_size; (void)ws_size;
    const float* X    = (const float*)d_in[0];
    const float* prob = (const float*)d_in[1];
    const float* W1   = (const float*)d_in[2];
    const float* b1   = (const float*)d_in[3];
    const float* W2   = (const float*)d_in[4];
    const float* b2   = (const float*)d_in[5];
    const int*   idx  = (const int*)d_in[6];
    float* out = (float*)d_out;
    int*   wsI = (int*)d_ws;
    float* oscale = (float*)(wsI + WS_OSCALE);

    moe_hist_kernel<<<NHBLK, 256, 0, stream>>>(idx, wsI);
    moe_prefix_kernel<<<1, 64, 0, stream>>>(wsI, out);
    moe_ginit_kernel<<<(MAXPAD + 255) / 256, 256, 0, stream>>>(wsI);
    moe_rank_kernel<<<NTOK / 256, 256, 0, stream>>>(idx, prob, wsI, oscale);
    moe_gemm_kernel<<<MAXTIL, 256, 0, stream>>>(X, prob, W1, b1, W2, b2,
                                                out, wsI, oscale);
}